// QuarticSSA_36369783062857
// MI455X (gfx1250) — compile-verified
//
#include <hip/hip_runtime.h>
#include <hip/hip_bf16.h>
#include <math.h>

typedef __attribute__((ext_vector_type(2))) float v2f;
typedef __attribute__((ext_vector_type(8))) float v8f;

#define B_    4
#define N_    4096
#define C_    128
#define S_    1024
#define K_    32
#define COUT_ 256
#define DPAD  144
#define BS    (B_ * S_)   // 4096

// d_out float offsets (tuple return, flattened in order)
#define OFF_CXYZ 0
#define OFF_CMAD (OFF_CXYZ + BS * 3)
#define OFF_CADJ (OFF_CMAD + BS * 3)
#define OFF_CPT  (OFF_CADJ + BS * 2)
#define OFF_O0   (OFF_CPT + BS * 4)   // 49152

// ---------------------------------------------------------------------------
// Farthest point sampling: 1 block per batch, sequential 1024 steps.
// Matches lax.scan semantics: output[i] is the center used at step i (idx 0 first).
// ---------------------------------------------------------------------------
__global__ void fps_kernel(const float* __restrict__ xyz, int* __restrict__ fps_idx) {
    const int b = blockIdx.x;
    const int tid = threadIdx.x;       // 256
    __shared__ float dist[N_];
    __shared__ float rv[256];
    __shared__ int ri[256];
    __shared__ float cen[3];
    __shared__ int s_far;
    const float* X = xyz + (size_t)b * N_ * 3;
    for (int i = tid; i < N_; i += 256) dist[i] = 1e10f;
    if (tid == 0) s_far = 0;
    __syncthreads();
    for (int it = 0; it < S_; ++it) {
        int far = s_far;
        if (tid == 0) {
            fps_idx[b * S_ + it] = far;
            cen[0] = X[far * 3 + 0];
            cen[1] = X[far * 3 + 1];
            cen[2] = X[far * 3 + 2];
        }
        __syncthreads();
        float bm = -1.0f;
        int bi = 0x7fffffff;
        for (int i = tid; i < N_; i += 256) {
            float dx = X[i * 3 + 0] - cen[0];
            float dy = X[i * 3 + 1] - cen[1];
            float dz = X[i * 3 + 2] - cen[2];
            float d = dx * dx + dy * dy + dz * dz;
            float nd = fminf(dist[i], d);
            dist[i] = nd;
            if (nd > bm || (nd == bm && i < bi)) { bm = nd; bi = i; }
        }
        rv[tid] = bm; ri[tid] = bi;
        __syncthreads();
        for (int s2 = 128; s2 > 0; s2 >>= 1) {
            if (tid < s2) {
                float ov = rv[tid + s2]; int oi = ri[tid + s2];
                if (ov > rv[tid] || (ov == rv[tid] && oi < ri[tid])) { rv[tid] = ov; ri[tid] = oi; }
            }
            __syncthreads();
        }
        if (tid == 0) s_far = ri[0];
        __syncthreads();
    }
}

// ---------------------------------------------------------------------------
// kNN (32 smallest, ascending, lowest-index tiebreak) only for the 4096
// selected centers. 1 block per (b,s); distance row lives in LDS.
// Uses the reference's sq_n + sq_m - 2*dot formula.
// ---------------------------------------------------------------------------
__global__ void knn_kernel(const float* __restrict__ xyz, const int* __restrict__ fps_idx,
                           int* __restrict__ knn_idx) {
    const int r = blockIdx.x;           // 0..4095
    const int b = r >> 10;
    const int tid = threadIdx.x;        // 256
    __shared__ float dist[N_];
    __shared__ float rv[256];
    __shared__ int ri[256];
    const float* X = xyz + (size_t)b * N_ * 3;
    const int n = fps_idx[r];
    const float cx = X[n * 3 + 0], cy = X[n * 3 + 1], cz = X[n * 3 + 2];
    const float sqn = cx * cx + cy * cy + cz * cz;
    for (int i = tid; i < N_; i += 256) {
        float x = X[i * 3 + 0], y = X[i * 3 + 1], z = X[i * 3 + 2];
        float sqm = x * x + y * y + z * z;
        dist[i] = sqn + sqm - 2.0f * (cx * x + cy * y + cz * z);
    }
    __syncthreads();
    for (int j = 0; j < K_; ++j) {
        float bm = 1e30f;
        int bi = 0x7fffffff;
        for (int i = tid; i < N_; i += 256) {
            float v = dist[i];
            if (v < bm || (v == bm && i < bi)) { bm = v; bi = i; }
        }
        rv[tid] = bm; ri[tid] = bi;
        __syncthreads();
        for (int s2 = 128; s2 > 0; s2 >>= 1) {
            if (tid < s2) {
                float ov = rv[tid + s2]; int oi = ri[tid + s2];
                if (ov < rv[tid] || (ov == rv[tid] && oi < ri[tid])) { rv[tid] = ov; ri[tid] = oi; }
            }
            __syncthreads();
        }
        if (tid == 0) {
            knn_idx[(size_t)r * K_ + j] = ri[0];
            dist[ri[0]] = 1e30f;
        }
        __syncthreads();
    }
}

// ---------------------------------------------------------------------------
// Gather small center outputs (center_xyz/mad/adj/pt) straight into d_out.
// ---------------------------------------------------------------------------
__global__ void centers_kernel(const float* __restrict__ xyz, const float* __restrict__ mad,
                               const float* __restrict__ adj, const float* __restrict__ pt,
                               const int* __restrict__ fps_idx, float* __restrict__ out) {
    const int i = blockIdx.x * blockDim.x + threadIdx.x;
    if (i >= BS) return;
    const int b = i >> 10;
    const size_t nb = (size_t)b * N_ + fps_idx[i];
    out[OFF_CXYZ + i * 3 + 0] = xyz[nb * 3 + 0];
    out[OFF_CXYZ + i * 3 + 1] = xyz[nb * 3 + 1];
    out[OFF_CXYZ + i * 3 + 2] = xyz[nb * 3 + 2];
    out[OFF_CMAD + i * 3 + 0] = mad[nb * 3 + 0];
    out[OFF_CMAD + i * 3 + 1] = mad[nb * 3 + 1];
    out[OFF_CMAD + i * 3 + 2] = mad[nb * 3 + 2];
    out[OFF_CADJ + i * 2 + 0] = adj[nb * 2 + 0];
    out[OFF_CADJ + i * 2 + 1] = adj[nb * 2 + 1];
    out[OFF_CPT + i * 4 + 0] = pt[nb * 4 + 0];
    out[OFF_CPT + i * 4 + 1] = pt[nb * 4 + 1];
    out[OFF_CPT + i * 4 + 2] = pt[nb * 4 + 2];
    out[OFF_CPT + i * 4 + 3] = pt[nb * 4 + 3];
}

// ---------------------------------------------------------------------------
// Zero-pad a [D, 256] weight matrix to [DPAD, 256] so GEMM inner loops can
// issue unconditional loads (no exec-mask branching around vmem).
// ---------------------------------------------------------------------------
__global__ void pad_weight_kernel(const float* __restrict__ W, int D, float* __restrict__ Wp) {
    const int i = blockIdx.x * blockDim.x + threadIdx.x;   // over DPAD*COUT
    if (i >= DPAD * COUT_) return;
    const int d = i / COUT_;
    Wp[i] = (d < D) ? W[i] : 0.0f;
}

// ---------------------------------------------------------------------------
// WMMA f32 16x16x4 GEMMs. One wave per 16x16 output tile, 8 waves/block.
// A fragment: lane l=lane&15 -> row m0+l; half=lane>>4 -> K pair k0+2*half.
// B fragment: col n0+l; same K pair. C/D: vgpr r -> row m0+r+8*half.
// All inner-loop loads are unconditional (operands pre-padded with zeros).
// ---------------------------------------------------------------------------

// Mqk[128, DPAD] = Wq[128,256] * Wkp[DPAD,256]^T  (Wkp rows >= D are zero)
__global__ void gemm_wqwkT_kernel(const float* __restrict__ Wq, const float* __restrict__ Wkp,
                                  float* __restrict__ Mqk) {
    const int wave = threadIdx.x >> 5, lane = threadIdx.x & 31;
    const int half = lane >> 4, l = lane & 15;
    const int Nt = DPAD / 16;   // 9
    const int tile = blockIdx.x * 8 + wave;
    if (tile >= (C_ / 16) * Nt) return;   // 72 tiles
    const int m0 = (tile / Nt) * 16, n0 = (tile % Nt) * 16;
    const float* arow = Wq + (size_t)(m0 + l) * COUT_;
    const float* brow = Wkp + (size_t)(n0 + l) * COUT_;
    v8f acc = {};
    for (int k0 = 0; k0 < COUT_; k0 += 4) {
        const int k = k0 + 2 * half;
        v2f a; a.x = arow[k]; a.y = arow[k + 1];
        v2f bb; bb.x = brow[k]; bb.y = brow[k + 1];
        acc = __builtin_amdgcn_wmma_f32_16x16x4_f32(false, a, false, bb, (short)0, acc, false, false);
    }
#pragma unroll
    for (int r2 = 0; r2 < 8; ++r2) {
        Mqk[(size_t)(m0 + r2 + 8 * half) * DPAD + n0 + l] = acc[r2];
    }
}

// t[4096, DPAD] = gather(fea, fps_idx)[4096,128] * Mqk[128, DPAD]
__global__ void gemm_t_kernel(const float* __restrict__ fea, const int* __restrict__ fps_idx,
                              const float* __restrict__ Mqk, float* __restrict__ t) {
    const int wave = threadIdx.x >> 5, lane = threadIdx.x & 31;
    const int half = lane >> 4, l = lane & 15;
    const int Nt = DPAD / 16;   // 9
    const int tile = blockIdx.x * 8 + wave;
    if (tile >= (BS / 16) * Nt) return;   // 2304 tiles
    const int m0 = (tile / Nt) * 16, n0 = (tile % Nt) * 16;
    const int m = m0 + l;
    const int b = m >> 10;
    const float* arow = fea + ((size_t)b * N_ + fps_idx[m]) * C_;
    v8f acc = {};
    for (int k0 = 0; k0 < C_; k0 += 4) {
        const int k = k0 + 2 * half;
        v2f a; a.x = arow[k]; a.y = arow[k + 1];
        v2f bb;
        bb.x = Mqk[(size_t)k * DPAD + n0 + l];
        bb.y = Mqk[(size_t)(k + 1) * DPAD + n0 + l];
        acc = __builtin_amdgcn_wmma_f32_16x16x4_f32(false, a, false, bb, (short)0, acc, false, false);
    }
#pragma unroll
    for (int r2 = 0; r2 < 8; ++r2) {
        t[(size_t)(m0 + r2 + 8 * half) * DPAD + n0 + l] = acc[r2];
    }
}

// O[4096,256] = wsum[4096, DPAD] * Wvp[DPAD,256]  (pad rows/cols are zero)
__global__ void gemm_out_kernel(const float* __restrict__ wsum, const float* __restrict__ Wvp,
                                float* __restrict__ O) {
    const int wave = threadIdx.x >> 5, lane = threadIdx.x & 31;
    const int half = lane >> 4, l = lane & 15;
    const int Nt = COUT_ / 16;  // 16
    const int tile = blockIdx.x * 8 + wave;
    if (tile >= (BS / 16) * Nt) return;   // 4096 tiles
    const int m0 = (tile / Nt) * 16, n0 = (tile % Nt) * 16;
    const float* arow = wsum + (size_t)(m0 + l) * DPAD;
    const float* bcol = Wvp + n0 + l;
    v8f acc = {};
    for (int k0 = 0; k0 < DPAD; k0 += 4) {
        const int k = k0 + 2 * half;
        v2f a; a.x = arow[k]; a.y = arow[k + 1];
        v2f bb;
        bb.x = bcol[(size_t)k * COUT_];
        bb.y = bcol[(size_t)(k + 1) * COUT_];
        acc = __builtin_amdgcn_wmma_f32_16x16x4_f32(false, a, false, bb, (short)0, acc, false, false);
    }
#pragma unroll
    for (int r2 = 0; r2 < 8; ++r2) {
        O[(size_t)(m0 + r2 + 8 * half) * COUT_ + n0 + l] = acc[r2];
    }
}

// ---------------------------------------------------------------------------
// Fused scores + softmax + weighted-sum. One wave per center; lane = neighbor.
// mode: 0=mad(D=131) 1=adj(132) 2=pt(136) 3=cst(131). Writes wsum[r, 0..144)
// with pad zeroed. Gathers raw features on the fly (L2-resident).
// ---------------------------------------------------------------------------
__global__ void attn_kernel(const float* __restrict__ fea,
                            const float* __restrict__ xyz, const float* __restrict__ mad,
                            const float* __restrict__ adj, const float* __restrict__ pt,
                            const int* __restrict__ fps_idx, const int* __restrict__ knn_idx,
                            const float* __restrict__ t, float* __restrict__ wsum,
                            int mode, int D) {
    __shared__ float tsh[8][DPAD];
    const int wave = threadIdx.x >> 5, lane = threadIdx.x & 31;
    const int r = blockIdx.x * 8 + wave;    // < 4096 with grid 512
    const int b = r >> 10;
    // stage t row into LDS (shared by all lanes of this wave)
    const float* trow = t + (size_t)r * DPAD;
    for (int d = lane; d < DPAD; d += 32) tsh[wave][d] = trow[d];
    __syncthreads();
    const float* tl = tsh[wave];

    const int cen = fps_idx[r];
    const size_t cb = (size_t)b * N_ + cen;
    float ex_c[4] = {0.f, 0.f, 0.f, 0.f};
    if (mode == 0)      { ex_c[0] = mad[cb * 3]; ex_c[1] = mad[cb * 3 + 1]; ex_c[2] = mad[cb * 3 + 2]; }
    else if (mode == 1) { ex_c[0] = adj[cb * 2]; ex_c[1] = adj[cb * 2 + 1]; }
    else if (mode == 2) { ex_c[0] = pt[cb * 4]; ex_c[1] = pt[cb * 4 + 1]; ex_c[2] = pt[cb * 4 + 2]; ex_c[3] = pt[cb * 4 + 3]; }
    else                { ex_c[0] = xyz[cb * 3]; ex_c[1] = xyz[cb * 3 + 1]; ex_c[2] = xyz[cb * 3 + 2]; }

    // score for this lane's neighbor
    const int nb = knn_idx[(size_t)r * K_ + lane];
    const size_t gb = (size_t)b * N_ + nb;
    const float* frow = fea + gb * C_;
    float sc = 0.0f;
#pragma unroll 8
    for (int c = 0; c < C_; ++c) sc += tl[c] * frow[c];
    if (mode == 0) {
        for (int j = 0; j < 3; ++j) sc += tl[C_ + j] * (mad[gb * 3 + j] - ex_c[j]);
    } else if (mode == 1) {
        sc += tl[C_ + 0] * adj[gb * 2 + 0] + tl[C_ + 1] * adj[gb * 2 + 1]
            + tl[C_ + 2] * ex_c[0] + tl[C_ + 3] * ex_c[1];
    } else if (mode == 2) {
        for (int j = 0; j < 4; ++j) sc += tl[C_ + j] * pt[gb * 4 + j];
        for (int j = 0; j < 4; ++j) sc += tl[C_ + 4 + j] * ex_c[j];
    } else {
        for (int j = 0; j < 3; ++j) sc += tl[C_ + j] * (xyz[gb * 3 + j] - ex_c[j]);
    }
    sc *= (1.0f / 16.0f);   // 1/sqrt(COUT)

    // wave32 softmax over the 32 neighbors
    float mx = sc;
    for (int off = 16; off > 0; off >>= 1) mx = fmaxf(mx, __shfl_xor(mx, off, 32));
    float e = expf(sc - mx);
    float ssum = e;
    for (int off = 16; off > 0; off >>= 1) ssum += __shfl_xor(ssum, off, 32);
    const float attn = e / ssum;

    // weighted sum over neighbors: lane owns dims lane, lane+32, lane+64, lane+96
    float a0 = 0.f, a1 = 0.f, a2 = 0.f, a3 = 0.f, eacc = 0.f;
    const int Dex = D - C_;   // 3/4/8/3
    for (int k = 0; k < K_; ++k) {
        const float ak = __shfl(attn, k, 32);
        const int nk = __shfl(nb, k, 32);
        const size_t gk = (size_t)b * N_ + nk;
        const float* fr = fea + gk * C_;
        a0 += ak * fr[lane];
        a1 += ak * fr[lane + 32];
        a2 += ak * fr[lane + 64];
        a3 += ak * fr[lane + 96];
        if (lane < Dex) {
            float ev;
            if (mode == 0)      ev = mad[gk * 3 + lane] - ex_c[lane];
            else if (mode == 1) ev = (lane < 2) ? adj[gk * 2 + lane] : ex_c[lane - 2];
            else if (mode == 2) ev = (lane < 4) ? pt[gk * 4 + lane] : ex_c[lane - 4];
            else                ev = xyz[gk * 3 + lane] - ex_c[lane];
            eacc += ak * ev;
        }
    }
    float* wr = wsum + (size_t)r * DPAD;
    wr[lane] = a0;
    wr[lane + 32] = a1;
    wr[lane + 64] = a2;
    wr[lane + 96] = a3;
    if (lane < 16) wr[C_ + lane] = (lane < Dex) ? eacc : 0.0f;   // zero the pad
}

// ---------------------------------------------------------------------------
extern "C" void kernel_launch(void* const* d_in, const int* in_sizes, int n_in,
                              void* d_out, int out_size, void* d_ws, size_t ws_size,
                              hipStream_t stream) {
    (void)in_sizes; (void)n_in; (void)out_size; (void)ws_size;
    const float* xyz = (const float*)d_in[0];
    const float* mad = (const float*)d_in[1];
    const float* adj = (const float*)d_in[2];
    const float* pt  = (const float*)d_in[3];
    const float* fea[4] = {(const float*)d_in[4], (const float*)d_in[5],
                           (const float*)d_in[6], (const float*)d_in[7]};
    const float* wq[4] = {(const float*)d_in[8],  (const float*)d_in[11],
                          (const float*)d_in[14], (const float*)d_in[17]};
    const float* wk[4] = {(const float*)d_in[9],  (const float*)d_in[12],
                          (const float*)d_in[15], (const float*)d_in[18]};
    const float* wv[4] = {(const float*)d_in[10], (const float*)d_in[13],
                          (const float*)d_in[16], (const float*)d_in[19]};
    const int Ds[4] = {C_ + 3, C_ + 4, C_ + 8, C_ + 3};
    float* out = (float*)d_out;

    // workspace layout (~5.6 MB)
    int* fps_idx = (int*)d_ws;                             // 4096 ints
    int* knn_idx = fps_idx + BS;                           // 4096*32 ints
    float* tbuf = (float*)(knn_idx + (size_t)BS * K_);     // 4096*144 f
    float* wsum = tbuf + (size_t)BS * DPAD;                // 4096*144 f
    float* Mqk  = wsum + (size_t)BS * DPAD;                // 128*144 f
    float* Wkp  = Mqk + (size_t)C_ * DPAD;                 // 144*256 f
    float* Wvp  = Wkp + (size_t)DPAD * COUT_;              // 144*256 f

    fps_kernel<<<B_, 256, 0, stream>>>(xyz, fps_idx);
    knn_kernel<<<BS, 256, 0, stream>>>(xyz, fps_idx, knn_idx);
    centers_kernel<<<BS / 256, 256, 0, stream>>>(xyz, mad, adj, pt, fps_idx, out);

    const int padBlocks = (DPAD * COUT_ + 255) / 256;   // 144
    for (int a = 0; a < 4; ++a) {
        pad_weight_kernel<<<padBlocks, 256, 0, stream>>>(wk[a], Ds[a], Wkp);
        pad_weight_kernel<<<padBlocks, 256, 0, stream>>>(wv[a], Ds[a], Wvp);
        gemm_wqwkT_kernel<<<9, 256, 0, stream>>>(wq[a], Wkp, Mqk);
        gemm_t_kernel<<<288, 256, 0, stream>>>(fea[a], fps_idx, Mqk, tbuf);
        attn_kernel<<<512, 256, 0, stream>>>(fea[a], xyz, mad, adj, pt,
                                             fps_idx, knn_idx, tbuf, wsum, a, Ds[a]);
        gemm_out_kernel<<<512, 256, 0, stream>>>(wsum, Wvp,
                                                 out + OFF_O0 + (size_t)a * BS * COUT_);
    }
}